// QuantumFeatureMap_23115513987347
// MI455X (gfx1250) — compile-verified
//
#include <hip/hip_runtime.h>
#include <math.h>

typedef __attribute__((ext_vector_type(2))) float v2f;
typedef __attribute__((ext_vector_type(8))) float v8f;

// Effective ±1 observable matrix after folding the CNOT ring (GF(2)-linear
// permutation) into the diagonal Z / ZZ observables. Output n has s-bit parity
// mask m_n (qubit 0 = MSB of state index s):
//   n : 0   1   2   3   4   5   6   7   8   9
//   m : 7  12  14  15  11   9   8   2   3   1
// packed 4 bits per output:
#define OBS_MASKS 0x13289BFEC7ull

__device__ __forceinline__ float pm_sign(unsigned mask, unsigned s) {
    return (__popc(mask & s) & 1) ? -1.0f : 1.0f;
}

__global__ __launch_bounds__(256) void QuantumFeatureMap_23115513987347_kernel(
        const float* __restrict__ x, float* __restrict__ out,
        int ntiles, int B) {
    const int lane = (int)(threadIdx.x & 31u);
    const int tile = (int)blockIdx.x * 8 + (int)(threadIdx.x >> 5u);
    if (tile >= ntiles) return;                 // wave-uniform: EXEC stays all-1s

    const int nb  = lane & 15;                  // batch-in-tile (B/D column) AND A row (output idx)
    const int h   = lane >> 4;                  // lane half selects K sub-pair
    const int row = tile * 16 + nb;             // global batch row
    const int rowc = row < B ? row : B - 1;     // safe clamp (B divisible by 16 in practice)

    // ---- B operand: per-row state probabilities (product form) ------------
    const float4 xv = *(const float4*)(x + (size_t)rowc * 4);
    const float c0 = cosf(1.57f * xv.x);
    const float c1 = cosf(1.57f * xv.y);
    const float c2 = cosf(1.57f * xv.z);
    const float c3 = cosf(1.57f * xv.w);
    const float p0 = 0.5f * (1.0f + c0), q0 = 0.5f * (1.0f - c0);
    const float p1 = 0.5f * (1.0f + c1), q1 = 0.5f * (1.0f - c1);
    const float p2 = 0.5f * (1.0f + c2), q2 = 0.5f * (1.0f - c2);
    const float p3 = 0.5f * (1.0f + c3), q3 = 0.5f * (1.0f - c3);

    // state s = 4k + 2h + j  ->  bits (b0,b1) = k, b2 = h, b3 = j
    const float f01[4] = { p0 * p1, p0 * q1, q0 * p1, q0 * q1 };
    const float P2h  = h ? q2 : p2;
    const float f23a = P2h * p3;                // j = 0
    const float f23b = P2h * q3;                // j = 1

    // ---- A operand: constant ±1 matrix row nb ------------------------------
    const bool valid = nb < 10;
    const unsigned mask = valid ? (unsigned)((OBS_MASKS >> (4 * nb)) & 0xFull) : 0u;

    // ---- D[n,b] = sum_s M[n,s] * probs[b,s] via 4 chained f32 WMMAs --------
    v8f acc = {0.0f, 0.0f, 0.0f, 0.0f, 0.0f, 0.0f, 0.0f, 0.0f};
#pragma unroll
    for (int k = 0; k < 4; ++k) {
        const unsigned s0 = 4u * (unsigned)k + 2u * (unsigned)h;
        v2f a, b;
        a[0] = valid ? pm_sign(mask, s0)      : 0.0f;
        a[1] = valid ? pm_sign(mask, s0 + 1u) : 0.0f;
        b[0] = f01[k] * f23a;
        b[1] = f01[k] * f23b;
        // (neg_a, A, neg_b, B, c_mod, C, reuse_a, reuse_b)
        acc = __builtin_amdgcn_wmma_f32_16x16x4_f32(false, a, false, b,
                                                    (short)0, acc, false, false);
    }

    // ---- Store: lane h=0 holds outputs 0..7 of row nb, h=1 holds 8..9 ------
    if (row < B) {
        float* o = out + (size_t)row * 10;      // 8-byte aligned (40B rows)
        if (h == 0) {
            *(float2*)(o + 0) = make_float2(acc[0], acc[1]);
            *(float2*)(o + 2) = make_float2(acc[2], acc[3]);
            *(float2*)(o + 4) = make_float2(acc[4], acc[5]);
            *(float2*)(o + 6) = make_float2(acc[6], acc[7]);
        } else {
            *(float2*)(o + 8) = make_float2(acc[0], acc[1]);
        }
    }
}

extern "C" void kernel_launch(void* const* d_in, const int* in_sizes, int n_in,
                              void* d_out, int out_size, void* d_ws, size_t ws_size,
                              hipStream_t stream) {
    const float* x = (const float*)d_in[0];
    float* out = (float*)d_out;

    const int B = in_sizes[0] / 4;              // rows (N_QUBITS = 4)
    const int ntiles = (B + 15) / 16;           // one 16-row tile per wave
    const int waves_per_block = 8;              // 256 threads = 8 wave32
    const int blocks = (ntiles + waves_per_block - 1) / waves_per_block;

    QuantumFeatureMap_23115513987347_kernel<<<blocks, 256, 0, stream>>>(x, out, ntiles, B);
}